// HypersphereLoss_6167573037472
// MI455X (gfx1250) — compile-verified
//
#include <hip/hip_runtime.h>

// HypersphereLoss, MI455X/gfx1250.
// One-pass formulation:  sum_{i in c} ||z_i - mu_c||^2 = S2_c - ||sums_c||^2 / cnt_c
// => single 512MB stream over z (~22us HBM floor at 23.3 TB/s).
//  - LDS-privatized per-class accumulators (52KB of 320KB/WGP), ds_add_f32
//    atomics made bank-conflict-free via an address-only XOR swizzle.
//  - wave32 __shfl_xor row reduction.
//  - finalize: v_wmma_f32_16x16x4_f32 Gram diagonal for ||sums_c||^2,
//    padded to 128 classes so all WMMA operand loads are branch-free.

typedef float v2f __attribute__((ext_vector_type(2)));
typedef float v4f __attribute__((ext_vector_type(4)));
typedef float v8f __attribute__((ext_vector_type(8)));

#define DD 128                    // feature dim
#define CC 100                    // num classes (fixed by reference)
#define CP 128                    // padded classes (8 WMMA tiles of 16)
#define LACCN (CC * DD + 2 * CC)  // LDS accum: 13000 floats (52000 B)
#define GACCN (CP * DD + 2 * CP)  // global accum (padded): 16640 floats

// Bank-conflict-free physical index for element d of a class row.
// Lanes 0-15 handle d<64 (bank class k mod 4); lanes 16-31 handle d>=64 and
// get XOR'ed to bank class (k^2) mod 4 -> 32 distinct banks per ds_add_f32.
__device__ __forceinline__ int swz(int d) { return d ^ ((d >> 5) & 2); }

__global__ __launch_bounds__(256) void hs_init(float* __restrict__ ws, int n) {
  int i = blockIdx.x * blockDim.x + threadIdx.x;
  if (i < n) ws[i] = 0.0f;
}

// One wave per row: lane handles d = 4*lane..4*lane+3 via a 16B load
// (global_load_b128, 512B per wave, fully coalesced, non-temporal).
__global__ __launch_bounds__(256) void hs_accum(const v4f* __restrict__ z4,
                                                const int* __restrict__ labels,
                                                float* __restrict__ gacc,
                                                int nrows) {
  __shared__ float lacc[LACCN];
  const int tid = threadIdx.x;
  for (int i = tid; i < LACCN; i += 256) lacc[i] = 0.0f;
  __syncthreads();

  float* lsums = lacc;                 // [CC*DD], XOR-swizzled layout
  float* lcnt  = lacc + CC * DD;       // [CC]
  float* ls2   = lacc + CC * DD + CC;  // [CC]

  const int lane   = tid & 31;
  const int wave   = (blockIdx.x * 256 + tid) >> 5;
  const int nwaves = (gridDim.x * 256) >> 5;
  const int d0     = lane * 4;         // this lane's first feature index
  const int p0     = swz(d0);          // swizzled base (d0..d0+3 share the swizzle)

  for (int r = wave; r < nrows; r += nwaves) {
    v4f v  = __builtin_nontemporal_load(&z4[r * (DD / 4) + lane]);
    int lab = __builtin_nontemporal_load(&labels[r]);   // uniform across wave
    if ((unsigned)lab < (unsigned)CC) {
      float* dst = lsums + lab * DD + p0;
      atomicAdd(dst + 0, v.x);                // ds_add_f32, conflict-free
      atomicAdd(dst + 1, v.y);
      atomicAdd(dst + 2, v.z);
      atomicAdd(dst + 3, v.w);
      float sq = v.x * v.x + v.y * v.y + v.z * v.z + v.w * v.w;
      #pragma unroll
      for (int off = 16; off; off >>= 1) sq += __shfl_xor(sq, off, 32);  // wave32
      if (lane == 0) {
        atomicAdd(&lcnt[lab], 1.0f);
        atomicAdd(&ls2[lab], sq);
      }
    }
  }
  __syncthreads();

  // Flush: unswizzle into linear padded global accumulators.
  for (int i = tid; i < CC * DD; i += 256) {
    int c = i >> 7, d = i & 127;
    float val = lsums[(c << 7) + swz(d)];
    if (val != 0.0f) atomicAdd(&gacc[i], val);     // global_atomic_add_f32
  }
  for (int c = tid; c < CC; c += 256) {
    float cv = lcnt[c], sv = ls2[c];
    if (cv != 0.0f) atomicAdd(&gacc[CP * DD + c], cv);
    if (sv != 0.0f) atomicAdd(&gacc[CP * DD + CP + c], sv);
  }
}

// Finalize (single block, 8 waves):
//  - centers[c][d] = sums/max(cnt,1)                    -> out[1 .. C*D]
//  - ||sums_c||^2 = diag(sums * sums^T) via V_WMMA_F32_16X16X4_F32,
//    one wave per 16-class tile (classes padded to 128 -> branch-free loads),
//    K reduced over DD in chunks of 4.
//  - loss = mean over present classes of (S2_c - ||sums_c||^2/cnt)/cnt -> out[0]
__global__ __launch_bounds__(256) void hs_finalize(const float* __restrict__ gacc,
                                                   float* __restrict__ out) {
  const float* gsums = gacc;
  const float* gcnt  = gacc + CP * DD;
  const float* gs2   = gacc + CP * DD + CP;

  __shared__ float sh[2];  // [0]=loss sum, [1]=n_present
  if (threadIdx.x < 2) sh[threadIdx.x] = 0.0f;
  __syncthreads();

  for (int i = threadIdx.x; i < CC * DD; i += 256) {
    int c = i / DD;
    out[1 + i] = gsums[i] / fmaxf(gcnt[c], 1.0f);
  }

  // --- WMMA Gram diagonal (branch-free: crow < CP always) ---
  const int lane = threadIdx.x & 31;
  const int wave = threadIdx.x >> 5;
  const int c0   = wave * 16;       // class tile base (0..112)
  const int half = lane >> 4;       // selects K pair within the 4-chunk
  const int l    = lane & 15;       // A row == B col for the symmetric product
  const float* row = gsums + (c0 + l) * DD;

  v8f acc = {};
  for (int k0 = 0; k0 < DD; k0 += 4) {
    v2f ab;
    ab.x = row[k0 + 2 * half];
    ab.y = row[k0 + 2 * half + 1];
    // A(16x4) and B(4x16) register images coincide for sums * sums^T.
    acc = __builtin_amdgcn_wmma_f32_16x16x4_f32(
        /*neg_a=*/false, ab, /*neg_b=*/false, ab,
        /*c_mod=*/(short)0, acc, /*reuse_a=*/false, /*reuse_b=*/false);
  }

  // C/D layout: VGPR r -> M=r (lanes 0-15), M=8+r (lanes 16-31), N=lane%16.
  // Diagonal: lanes 0..7 -> acc[lane]; lanes 24..31 -> acc[lane-24].
  float d8[8];
  #pragma unroll
  for (int r = 0; r < 8; ++r) d8[r] = acc[r];

  int c = -1;
  float diag = 0.0f;
  if (lane < 8)        { c = c0 + lane;      diag = d8[lane]; }
  else if (lane >= 24) { c = c0 + lane - 16; diag = d8[lane - 24]; }

  if (c >= 0 && c < CC) {
    float cnt = gcnt[c];
    if (cnt > 0.0f) {
      float per = (gs2[c] - diag / cnt) / cnt;
      atomicAdd(&sh[0], per);
      atomicAdd(&sh[1], 1.0f);
    }
  }
  __syncthreads();
  if (threadIdx.x == 0) out[0] = sh[0] / fmaxf(sh[1], 1.0f);
}

extern "C" void kernel_launch(void* const* d_in, const int* in_sizes, int n_in,
                              void* d_out, int out_size, void* d_ws, size_t ws_size,
                              hipStream_t stream) {
  const float* z      = (const float*)d_in[0];
  const int*   labels = (const int*)d_in[1];   // labels per harness int convention
  float*       ws     = (float*)d_ws;          // 16640 floats of scratch
  float*       out    = (float*)d_out;         // [0]=loss, [1..]=centers (12801)
  (void)n_in; (void)out_size; (void)ws_size;

  const int nrows = in_sizes[0] / DD;          // N = 1,000,000

  hs_init<<<(GACCN + 255) / 256, 256, 0, stream>>>(ws, GACCN);
  hs_accum<<<512, 256, 0, stream>>>((const v4f*)z, labels, ws, nrows);
  hs_finalize<<<1, 256, 0, stream>>>(ws, out);
}